// Attention4NovelRec_59528246722885
// MI455X (gfx1250) — compile-verified
//
#include <hip/hip_runtime.h>
#include <hip/hip_bf16.h>
#include <math.h>

#define BATCH 1024
#define SEQ   49
#define DIM   128
#define HEADS 4
#define HDIM  32
#define VOCAB 100001
#define NEGV  (-1e8f)

typedef __attribute__((ext_vector_type(2))) float v2f;
typedef __attribute__((ext_vector_type(8))) float v8f;

// D = A(16x4) * B(4x16) + C(16x16), all fp32.
// A frag: lane L<16 -> {A[L][k], A[L][k+1]}, lane L>=16 -> {A[L-16][k+2], A[L-16][k+3]}
// B frag: lane L<16 -> {B[k][L], B[k+1][L]},  lane L>=16 -> {B[k+2][L-16], B[k+3][L-16]}
// C/D   : VGPR r, lanes 0-15 -> (M=r, N=lane), lanes 16-31 -> (M=r+8, N=lane-16)
__device__ __forceinline__ v8f wmma4(v2f a, v2f b, v8f c) {
  return __builtin_amdgcn_wmma_f32_16x16x4_f32(false, a, false, b, (short)0, c,
                                               false, false);
}

// ---------------------------------------------------------------------------
// Kernel A: x[b,s,:] = emb[seq] + subgenre[subtype] + PE(s)
// grid = B*S blocks of DIM threads
// ---------------------------------------------------------------------------
__global__ void embed_kernel(const int* __restrict__ seq,
                             const int* __restrict__ subtype,
                             const float* __restrict__ emb,
                             const float* __restrict__ sub,
                             float* __restrict__ x) {
  const int bs = blockIdx.x;            // [0, B*S)
  const int d  = threadIdx.x;           // [0, 128)
  const int s  = bs % SEQ;
  const int tok = seq[bs];
  const int st  = subtype[bs];
  const float pos = (float)s;
  const float pe  = (d & 1) ? cosf(pos) : sinf(pos);
  x[(size_t)bs * DIM + d] = emb[(size_t)tok * DIM + d] +
                            sub[(size_t)st * DIM + d] + pe;
}

// ---------------------------------------------------------------------------
// Kernel B: per batch element: QKV (WMMA) -> masked softmax attention ->
// output projection (WMMA) -> SelfEnc pooling.  1 block / batch element.
// 256 threads = 8 waves; wave w owns output columns [16w, 16w+16).
// S padded 49 -> 64 rows (pad rows zero).
// ---------------------------------------------------------------------------
__global__ __launch_bounds__(256) void attn_kernel(
    const float* __restrict__ x, const int* __restrict__ seq,
    const float* __restrict__ wq, const float* __restrict__ bq,
    const float* __restrict__ wk, const float* __restrict__ bk,
    const float* __restrict__ wvp, const float* __restrict__ bv,
    const float* __restrict__ wo, const float* __restrict__ bo,
    const float* __restrict__ w_enc, const float* __restrict__ b_enc,
    float* __restrict__ pooled) {
  __shared__ float xs[64][130];    // X, later reused as attention output (AV)
  __shared__ float qs[64][130];    // Q, later reused as ENC
  __shared__ float ks2[64][130];   // K
  __shared__ float vs[64][130];    // V
  __shared__ float sc[SEQ][66];    // per-head score/prob rows
  __shared__ float wrow[64];       // SelfEnc per-row weights
  __shared__ int   spad[64];       // padding flags

  const int b    = blockIdx.x;
  const int tid  = threadIdx.x;
  const int lane = tid & 31;
  const int wid  = tid >> 5;       // wave 0..7
  const int hlf  = lane >> 4;      // lane half
  const int l15  = lane & 15;
  const int col  = wid * 16 + l15; // this wave's output column in [0,128)

  // ---- load X (pad to 64 rows with zeros), padding flags ----
  const float* xg = x + (size_t)b * SEQ * DIM;
  for (int idx = tid; idx < 64 * DIM; idx += 256) {
    const int r = idx >> 7, c = idx & 127;
    xs[r][c] = (r < SEQ) ? xg[r * DIM + c] : 0.0f;
  }
  if (tid < 64) spad[tid] = (tid < SEQ) ? (seq[b * SEQ + tid] == 0) : 1;
  __syncthreads();

  // ---- fused Q/K/V projections: [64x128] @ [128x128], WMMA f32 16x16x4 ----
  v8f aq[4], ak[4], av4[4];
  for (int mt = 0; mt < 4; ++mt) { aq[mt] = (v8f){}; ak[mt] = (v8f){}; av4[mt] = (v8f){}; }
  for (int k4 = 0; k4 < DIM; k4 += 4) {
    const int kk = k4 + 2 * hlf;
    v2f bqf, bkf, bvf;
    bqf[0] = wq[kk * DIM + col];  bqf[1] = wq[(kk + 1) * DIM + col];
    bkf[0] = wk[kk * DIM + col];  bkf[1] = wk[(kk + 1) * DIM + col];
    bvf[0] = wvp[kk * DIM + col]; bvf[1] = wvp[(kk + 1) * DIM + col];
    for (int mt = 0; mt < 4; ++mt) {
      const int row = mt * 16 + l15;
      v2f a; a[0] = xs[row][kk]; a[1] = xs[row][kk + 1];
      aq[mt]  = wmma4(a, bqf, aq[mt]);
      ak[mt]  = wmma4(a, bkf, ak[mt]);
      av4[mt] = wmma4(a, bvf, av4[mt]);
    }
  }
  {
    const float bqv = bq[col], bkv = bk[col], bvv = bv[col];
    for (int mt = 0; mt < 4; ++mt)
      for (int r = 0; r < 8; ++r) {
        const int row = mt * 16 + r + 8 * hlf;
        qs[row][col]  = aq[mt][r] + bqv;
        ks2[row][col] = ak[mt][r] + bkv;
        vs[row][col]  = av4[mt][r] + bvv;
      }
  }
  __syncthreads();

  // ---- attention per head: scores -> softmax -> AV (into xs) ----
  const float inv_d = 1.0f / (float)HDIM;
  for (int h = 0; h < HEADS; ++h) {
    const int c0 = h * HDIM;
    for (int idx = tid; idx < SEQ * SEQ; idx += 256) {
      const int i = idx / SEQ, j = idx - i * SEQ;
      float dot = 0.0f;
      for (int c = 0; c < HDIM; ++c) dot += qs[i][c0 + c] * ks2[j][c0 + c];
      sc[i][j] = (spad[i] | spad[j]) ? NEGV : dot * inv_d;
    }
    __syncthreads();
    if (tid < SEQ) {
      float m = -INFINITY;
      for (int j = 0; j < SEQ; ++j) m = fmaxf(m, sc[tid][j]);
      float ssum = 0.0f;
      for (int j = 0; j < SEQ; ++j) { const float e = expf(sc[tid][j] - m); sc[tid][j] = e; ssum += e; }
      const float inv = 1.0f / ssum;
      for (int j = 0; j < SEQ; ++j) sc[tid][j] *= inv;
    }
    __syncthreads();
    for (int idx = tid; idx < SEQ * HDIM; idx += 256) {
      const int i = idx >> 5, c = idx & 31;
      float acc = 0.0f;
      for (int j = 0; j < SEQ; ++j) acc += sc[i][j] * vs[j][c0 + c];
      xs[i][c0 + c] = acc;   // rows >= SEQ stay zero from the X load
    }
    __syncthreads();
  }

  // ---- output projection: enc = AV @ wo + bo (into qs), WMMA ----
  v8f ae[4];
  for (int mt = 0; mt < 4; ++mt) ae[mt] = (v8f){};
  for (int k4 = 0; k4 < DIM; k4 += 4) {
    const int kk = k4 + 2 * hlf;
    v2f bf; bf[0] = wo[kk * DIM + col]; bf[1] = wo[(kk + 1) * DIM + col];
    for (int mt = 0; mt < 4; ++mt) {
      const int row = mt * 16 + l15;
      v2f a; a[0] = xs[row][kk]; a[1] = xs[row][kk + 1];
      ae[mt] = wmma4(a, bf, ae[mt]);
    }
  }
  __syncthreads();   // all waves done reading qs as Q before overwriting w/ enc
  {
    const float bov = bo[col];
    for (int mt = 0; mt < 4; ++mt)
      for (int r = 0; r < 8; ++r) {
        const int row = mt * 16 + r + 8 * hlf;
        qs[row][col] = ae[mt][r] + bov;
      }
  }
  __syncthreads();

  // ---- SelfEnc pooling: w[i] = enc[i].w_enc + b_enc; pooled = sum_i w[i]*enc[i]
  if (tid < SEQ) {
    float acc = b_enc[0];
    for (int d = 0; d < DIM; ++d) acc += qs[tid][d] * w_enc[d];
    wrow[tid] = acc;
  }
  __syncthreads();
  if (tid < DIM) {
    float p = 0.0f;
    for (int i = 0; i < SEQ; ++i) p += wrow[i] * qs[i][tid];
    pooled[(size_t)b * DIM + tid] = p;
  }
}

// ---------------------------------------------------------------------------
// Kernel C: out[1024, 100001] = pooled[1024,128] @ w_score[128,100001] + bias
// block = 256 thr (8 waves) -> M=128 rows x N=128 cols per block.
// Each wave: one 16-col strip x 8 M-tiles -> each w_score fragment load
// feeds 8 wmma (w_score HBM traffic ~= compulsory 51 MB; output write bound).
// A tile in LDS, stride 132: 16B-aligned rows for b128 copy, and the
// A-fragment b64 reads hit all 64 banks (4*l15 + k4 + {0..3}) conflict-free.
// ---------------------------------------------------------------------------
__global__ __launch_bounds__(256) void score_kernel(
    const float* __restrict__ pooled, const float* __restrict__ w_score,
    const float* __restrict__ b_score, float* __restrict__ out) {
  __shared__ float As[128][132];
  const int tid  = threadIdx.x;
  const int lane = tid & 31;
  const int wid  = tid >> 5;
  const int hlf  = lane >> 4;
  const int l15  = lane & 15;
  const int m0   = blockIdx.y * 128;
  const int n0   = blockIdx.x * 128 + wid * 16;

  // pooled[m0..m0+127][0..127] -> As, vectorized b128 (rows 16B-aligned)
  for (int idx = tid; idx < 128 * (DIM / 4); idx += 256) {
    const int r = idx >> 5, c4 = (idx & 31) * 4;
    *(float4*)&As[r][c4] =
        *(const float4*)&pooled[(size_t)(m0 + r) * DIM + c4];
  }
  __syncthreads();

  const int colr = n0 + l15;
  const int col  = (colr < VOCAB) ? colr : (VOCAB - 1);  // clamp loads; store guarded
  v8f acc[8];
  for (int mt = 0; mt < 8; ++mt) acc[mt] = (v8f){};
  for (int k4 = 0; k4 < DIM; k4 += 4) {
    const int kk = k4 + 2 * hlf;
    v2f bf; bf[0] = w_score[(size_t)kk * VOCAB + col];
            bf[1] = w_score[(size_t)(kk + 1) * VOCAB + col];
    for (int mt = 0; mt < 8; ++mt) {
      v2f a; a[0] = As[mt * 16 + l15][kk]; a[1] = As[mt * 16 + l15][kk + 1];
      acc[mt] = wmma4(a, bf, acc[mt]);
    }
  }
  if (colr < VOCAB) {
    const float bias = b_score[colr];
    for (int mt = 0; mt < 8; ++mt)
      for (int r = 0; r < 8; ++r) {
        const int row = m0 + mt * 16 + r + 8 * hlf;
        out[(size_t)row * VOCAB + colr] = acc[mt][r] + bias;
      }
  }
}

// ---------------------------------------------------------------------------
// Kernel D: scatter consumed mask. All writers store the same NEG value.
// ---------------------------------------------------------------------------
__global__ void mask_kernel(const int* __restrict__ seq, float* __restrict__ out) {
  const int t = blockIdx.x * 256 + threadIdx.x;
  if (t >= BATCH * SEQ) return;
  const int b = t / SEQ;
  const int v = seq[t];
  out[(size_t)b * VOCAB + v] = NEGV;
}

extern "C" void kernel_launch(void* const* d_in, const int* in_sizes, int n_in,
                              void* d_out, int out_size, void* d_ws, size_t ws_size,
                              hipStream_t stream) {
  (void)in_sizes; (void)n_in; (void)out_size; (void)ws_size;
  const int*   seq      = (const int*)d_in[0];
  const int*   subtype  = (const int*)d_in[1];
  const float* emb      = (const float*)d_in[2];
  const float* sub      = (const float*)d_in[3];
  const float* wq       = (const float*)d_in[4];
  const float* bq       = (const float*)d_in[5];
  const float* wk       = (const float*)d_in[6];
  const float* bk       = (const float*)d_in[7];
  const float* wvp      = (const float*)d_in[8];
  const float* bv       = (const float*)d_in[9];
  const float* wo       = (const float*)d_in[10];
  const float* bo       = (const float*)d_in[11];
  const float* w_enc    = (const float*)d_in[12];
  const float* b_enc    = (const float*)d_in[13];
  const float* w_score  = (const float*)d_in[14];
  const float* b_score  = (const float*)d_in[15];
  float* out = (float*)d_out;

  float* x      = (float*)d_ws;                       // B*S*D fp32 (25.7 MB)
  float* pooled = x + (size_t)BATCH * SEQ * DIM;      // B*D fp32 (0.5 MB)

  embed_kernel<<<BATCH * SEQ, DIM, 0, stream>>>(seq, subtype, emb, sub, x);
  attn_kernel<<<BATCH, 256, 0, stream>>>(x, seq, wq, bq, wk, bk, wvp, bv,
                                         wo, bo, w_enc, b_enc, pooled);
  dim3 gC((VOCAB + 127) / 128, BATCH / 128);
  score_kernel<<<gC, 256, 0, stream>>>(pooled, w_score, b_score, out);
  mask_kernel<<<(BATCH * SEQ + 255) / 256, 256, 0, stream>>>(seq, out);
}